// point_transformer_13443247637193
// MI455X (gfx1250) — compile-verified
//
#include <hip/hip_runtime.h>
#include <hip/hip_bf16.h>
#include <cstddef>

// Problem constants (from reference)
#define NPTS 50000
#define HNB  16
#define CIN  256
#define CCH  256
#define CRD  64
#define CPG  32
#define GRP  8

typedef __attribute__((ext_vector_type(2))) float v2f;
typedef __attribute__((ext_vector_type(8))) float v8f;

__device__ __forceinline__ float lrelu(float x) {
    return x > 0.f ? x : 0.1f * x;
}

// D = A(16x4 f32) * B(4x16 f32) + C, full-wave (wave32) matrix op
__device__ __forceinline__ v8f wmma4(v2f a, v2f b, v8f c) {
    return __builtin_amdgcn_wmma_f32_16x16x4_f32(
        /*neg_a=*/false, a, /*neg_b=*/false, b,
        /*c_mod=*/(short)0, c, /*reuse_a=*/false, /*reuse_b=*/false);
}

// ---------------------------------------------------------------------------
// Kernel 1: q/k/v = s_feats @ W{q,k,v} + b  — one 16-row tile of N per block,
// 8 waves cover the 48 output column tiles (3 matrices x 16 tiles).
// ---------------------------------------------------------------------------
__global__ void __launch_bounds__(256)
qkv_kernel(const float* __restrict__ s_feats,
           const float* __restrict__ Wq, const float* __restrict__ bq,
           const float* __restrict__ Wk, const float* __restrict__ bk,
           const float* __restrict__ Wv, const float* __restrict__ bv,
           float* __restrict__ qb, float* __restrict__ kb, float* __restrict__ vb) {
    __shared__ float As[16][260];   // stride 260: conflict-free fragment reads

    const int tid = threadIdx.x;
    const int n0  = blockIdx.x * 16;

    // Stage A tile (16 x 256 f32) into LDS, float4 coalesced.
#pragma unroll
    for (int t = 0; t < 4; ++t) {
        int f   = tid + t * 256;        // float4 index, 1024 total
        int row = f >> 6;               // 64 float4 per row
        int col = (f & 63) * 4;
        float4 a4 = *(const float4*)(s_feats + (size_t)(n0 + row) * CIN + col);
        As[row][col + 0] = a4.x; As[row][col + 1] = a4.y;
        As[row][col + 2] = a4.z; As[row][col + 3] = a4.w;
    }
    __syncthreads();

    const int lane  = tid & 31;
    const int wave  = tid >> 5;
    const int m     = lane & 15;
    const int kOff  = (lane >> 4) << 1;  // lanes 16-31 hold K+2, K+3
    const int mBase = (lane >> 4) << 3;  // lanes 16-31 hold M=8..15

    for (int j = 0; j < 6; ++j) {
        int t     = wave * 6 + j;        // 0..47
        int matId = t >> 4;
        int ct    = t & 15;
        const float* W; const float* bias; float* outp;
        if (matId == 0)      { W = Wq; bias = bq; outp = qb; }
        else if (matId == 1) { W = Wk; bias = bk; outp = kb; }
        else                 { W = Wv; bias = bv; outp = vb; }

        const int nn = ct * 16 + m;
        v8f acc = {};
        for (int k0 = 0; k0 < CIN; k0 += 4) {
            int kk = k0 + kOff;
            v2f a; a.x = As[m][kk];          a.y = As[m][kk + 1];
            v2f b; b.x = W[kk * CCH + nn];   b.y = W[(kk + 1) * CCH + nn];
            acc = wmma4(a, b, acc);
        }
        float bval = bias[nn];
#pragma unroll
        for (int r = 0; r < 8; ++r)
            outp[(size_t)(n0 + mBase + r) * CCH + nn] = acc[r] + bval;
    }
}

// ---------------------------------------------------------------------------
// Kernel 2: one point per block. Gather neighbors, geom GEMM (fused epilogue
// producing a0 = lrelu(qf - nk - geom) and nvg = nv - geom in LDS), attention
// MLP GEMMs, softmax over H, weighted reduction to out.
// ---------------------------------------------------------------------------
__global__ void __launch_bounds__(256)
attn_kernel(const float* __restrict__ q_pts, const float* __restrict__ s_pts,
            const int*   __restrict__ neighb,
            const float* __restrict__ qb, const float* __restrict__ kb,
            const float* __restrict__ vb,
            const float* __restrict__ Wd1, const float* __restrict__ bd1,
            const float* __restrict__ Wd2, const float* __restrict__ bd2,
            const float* __restrict__ Wa1, const float* __restrict__ ba1,
            const float* __restrict__ Wa2, const float* __restrict__ ba2,
            float* __restrict__ out) {
    __shared__ float sNK[16][260];   // nk  -> a0 after geom epilogue
    __shared__ float sNV[16][260];   // nv  -> nv - geom after epilogue
    __shared__ float sH [16][68];    // lrelu(rel@Wd1+bd1)
    __shared__ float sQF[256];
    __shared__ float sA1[16][36];
    __shared__ float sA2[16][36];
    __shared__ float sAttn[16][32];
    __shared__ int   sNB[16];

    const int n   = blockIdx.x;
    const int tid = threadIdx.x;

    if (tid < 16) sNB[tid] = neighb[(size_t)n * HNB + tid];
    __syncthreads();

    // --- gather nk/nv rows, qf row, and compute hidden = lrelu(rel@Wd1+bd1)
    {
        const int h  = tid >> 4;
        const int c0 = (tid & 15) * 16;
        const int nb = sNB[h];
        const float* krow = kb + (size_t)nb * CCH;
        const float* vrow = vb + (size_t)nb * CCH;
#pragma unroll
        for (int i = 0; i < 16; i += 4) {
            float4 tk = *(const float4*)(krow + c0 + i);
            sNK[h][c0 + i] = tk.x; sNK[h][c0 + i + 1] = tk.y;
            sNK[h][c0 + i + 2] = tk.z; sNK[h][c0 + i + 3] = tk.w;
            float4 tv = *(const float4*)(vrow + c0 + i);
            sNV[h][c0 + i] = tv.x; sNV[h][c0 + i + 1] = tv.y;
            sNV[h][c0 + i + 2] = tv.z; sNV[h][c0 + i + 3] = tv.w;
        }
        sQF[tid] = qb[(size_t)sNB[0] * CCH + tid];

        const float rx = s_pts[(size_t)nb * 3 + 0] - q_pts[(size_t)n * 3 + 0];
        const float ry = s_pts[(size_t)nb * 3 + 1] - q_pts[(size_t)n * 3 + 1];
        const float rz = s_pts[(size_t)nb * 3 + 2] - q_pts[(size_t)n * 3 + 2];
        const int hc0 = (tid & 15) * 4;
#pragma unroll
        for (int j = 0; j < 4; ++j) {
            int c = hc0 + j;
            float hv = rx * Wd1[c] + ry * Wd1[CRD + c] + rz * Wd1[2 * CRD + c] + bd1[c];
            sH[h][c] = lrelu(hv);
        }
    }
    __syncthreads();

    const int lane  = tid & 31;
    const int wave  = tid >> 5;
    const int m     = lane & 15;
    const int kOff  = (lane >> 4) << 1;
    const int mBase = (lane >> 4) << 3;

    // --- geom = hidden @ Wd2 + bd2, fused epilogue: a0 & nvg in place
    for (int j = 0; j < 2; ++j) {
        const int ct = wave * 2 + j;
        const int nn = ct * 16 + m;
        v8f acc = {};
        for (int k0 = 0; k0 < CRD; k0 += 4) {
            int kk = k0 + kOff;
            v2f a; a.x = sH[m][kk];           a.y = sH[m][kk + 1];
            v2f b; b.x = Wd2[kk * CCH + nn];  b.y = Wd2[(kk + 1) * CCH + nn];
            acc = wmma4(a, b, acc);
        }
        const float bval = bd2[nn];
        const float qfv  = sQF[nn];
#pragma unroll
        for (int r = 0; r < 8; ++r) {
            int row = mBase + r;
            float g = acc[r] + bval;
            sNK[row][nn] = lrelu(qfv - sNK[row][nn] - g);  // a0
            sNV[row][nn] = sNV[row][nn] - g;               // nv - geom
        }
    }
    __syncthreads();

    // --- a1 = lrelu(a0 @ Wa1 + ba1)   (16 x 256) @ (256 x 32)
    if (wave < 2) {
        const int nn = wave * 16 + m;
        v8f acc = {};
        for (int k0 = 0; k0 < CCH; k0 += 4) {
            int kk = k0 + kOff;
            v2f a; a.x = sNK[m][kk];          a.y = sNK[m][kk + 1];
            v2f b; b.x = Wa1[kk * CPG + nn];  b.y = Wa1[(kk + 1) * CPG + nn];
            acc = wmma4(a, b, acc);
        }
        const float bval = ba1[nn];
#pragma unroll
        for (int r = 0; r < 8; ++r)
            sA1[mBase + r][nn] = lrelu(acc[r] + bval);
    }
    __syncthreads();

    // --- a2 = a1 @ Wa2 + ba2   (16 x 32) @ (32 x 32)
    if (wave < 2) {
        const int nn = wave * 16 + m;
        v8f acc = {};
        for (int k0 = 0; k0 < CPG; k0 += 4) {
            int kk = k0 + kOff;
            v2f a; a.x = sA1[m][kk];          a.y = sA1[m][kk + 1];
            v2f b; b.x = Wa2[kk * CPG + nn];  b.y = Wa2[(kk + 1) * CPG + nn];
            acc = wmma4(a, b, acc);
        }
        const float bval = ba2[nn];
#pragma unroll
        for (int r = 0; r < 8; ++r)
            sA2[mBase + r][nn] = acc[r] + bval;
    }
    __syncthreads();

    // --- softmax over H per cpg column
    if (tid < CPG) {
        float mx = -3.4e38f;
#pragma unroll
        for (int h = 0; h < HNB; ++h) mx = fmaxf(mx, sA2[h][tid]);
        float s = 0.f;
#pragma unroll
        for (int h = 0; h < HNB; ++h) {
            float e = __expf(sA2[h][tid] - mx);
            sAttn[h][tid] = e;
            s += e;
        }
        float inv = 1.f / s;
#pragma unroll
        for (int h = 0; h < HNB; ++h) sAttn[h][tid] *= inv;
    }
    __syncthreads();

    // --- out[n,c] = sum_h nvg[h][c] * attn[h][c/8]
    {
        float acc = 0.f;
        const int cg = tid >> 3;
#pragma unroll
        for (int h = 0; h < HNB; ++h)
            acc += sNV[h][tid] * sAttn[h][cg];
        out[(size_t)n * CCH + tid] = acc;
    }
}

// ---------------------------------------------------------------------------
extern "C" void kernel_launch(void* const* d_in, const int* in_sizes, int n_in,
                              void* d_out, int out_size, void* d_ws, size_t ws_size,
                              hipStream_t stream) {
    const float* q_pts   = (const float*)d_in[0];
    const float* s_pts   = (const float*)d_in[1];
    const float* s_feats = (const float*)d_in[2];
    const int*   neighb  = (const int*)d_in[3];
    const float* Wq = (const float*)d_in[4];  const float* bq = (const float*)d_in[5];
    const float* Wk = (const float*)d_in[6];  const float* bk = (const float*)d_in[7];
    const float* Wv = (const float*)d_in[8];  const float* bv = (const float*)d_in[9];
    const float* Wd1 = (const float*)d_in[10]; const float* bd1 = (const float*)d_in[11];
    const float* Wd2 = (const float*)d_in[12]; const float* bd2 = (const float*)d_in[13];
    const float* Wa1 = (const float*)d_in[14]; const float* ba1 = (const float*)d_in[15];
    const float* Wa2 = (const float*)d_in[16]; const float* ba2 = (const float*)d_in[17];
    float* out = (float*)d_out;

    float* qb = (float*)d_ws;                       // N*C floats
    float* kb = qb + (size_t)NPTS * CCH;            // N*C floats
    float* vb = kb + (size_t)NPTS * CCH;            // N*C floats

    qkv_kernel<<<NPTS / 16, 256, 0, stream>>>(s_feats, Wq, bq, Wk, bk, Wv, bv,
                                              qb, kb, vb);
    attn_kernel<<<NPTS, 256, 0, stream>>>(q_pts, s_pts, neighb, qb, kb, vb,
                                          Wd1, bd1, Wd2, bd2, Wa1, ba1, Wa2, ba2,
                                          out);
}